// ModelWrapper_66340064854351
// MI455X (gfx1250) — compile-verified
//
#include <hip/hip_runtime.h>

#define N_NODES  50000
#define N_EDGES  800000
#define DIM      256
#define N_GRAPHS 128
#define N_CLASSES 2

typedef __attribute__((ext_vector_type(2))) float v2f;
typedef __attribute__((ext_vector_type(8))) float v8f;

// ---------------------------------------------------------------- utilities
__global__ void k_fill(float* __restrict__ p, float v, int n) {
    int i = blockIdx.x * blockDim.x + threadIdx.x;
    if (i < n) p[i] = v;
}

// deg[dst] += ew  (deg pre-filled with 1.0 for the self-loop)
__global__ void k_edge_deg(const long long* __restrict__ ei,
                           const float* __restrict__ ew,
                           float* __restrict__ deg) {
    int e = blockIdx.x * blockDim.x + threadIdx.x;
    if (e < N_EDGES) {
        int d = (int)ei[N_EDGES + e];
        atomicAdd(&deg[d], ew[e]);
    }
}

__global__ void k_rsqrt_inplace(float* __restrict__ p, int n) {
    int i = blockIdx.x * blockDim.x + threadIdx.x;
    if (i < n) p[i] = rsqrtf(p[i]);   // deg >= 1.0 always (self loop)
}

// ---------------------------------------------------------------- fp32 WMMA GEMM
// C[M,256] = A[M,256] @ B[256,256], M = 50000 (divisible by 16).
// Block: 512 threads = 16 waves. Block owns 16 rows of A (staged in LDS),
// wave w owns output columns [16w, 16w+16).
__global__ __launch_bounds__(512)
void k_gemm_wmma_f32(const float* __restrict__ A,
                     const float* __restrict__ B,
                     float* __restrict__ C) {
    __shared__ float lds[16 * 257];          // pad stride 257 -> conflict-free column reads
    const int m0 = blockIdx.x * 16;

    // cooperative stage of the 16x256 A tile
    for (int idx = threadIdx.x; idx < 16 * 256; idx += 512) {
        int r = idx >> 8, c = idx & 255;
        lds[r * 257 + c] = A[(size_t)(m0 + r) * DIM + c];
    }
    __syncthreads();

    const int wave  = threadIdx.x >> 5;
    const int lane  = threadIdx.x & 31;
    const int n0    = wave * 16;
    const int row   = lane & 15;             // A-fragment row (M)
    const int khalf = (lane >> 4) * 2;       // lanes 16-31 hold K+2,K+3
    const int bcol  = n0 + (lane & 15);      // B-fragment column (N)

    v8f acc = {0.f, 0.f, 0.f, 0.f, 0.f, 0.f, 0.f, 0.f};

    #pragma unroll 4
    for (int k = 0; k < DIM; k += 4) {
        v2f a, b;
        a.x = lds[row * 257 + k + khalf];
        a.y = lds[row * 257 + k + khalf + 1];
        b.x = B[(size_t)(k + khalf) * DIM + bcol];
        b.y = B[(size_t)(k + khalf + 1) * DIM + bcol];
        // V_WMMA_F32_16X16X4_F32: D = A(16x4) x B(4x16) + C
        acc = __builtin_amdgcn_wmma_f32_16x16x4_f32(
                  false, a, false, b, (short)0, acc, false, false);
    }

    // C/D layout: VGPR r holds M = r (lanes 0-15) / M = 8+r (lanes 16-31), N = lane%16
    const int crow0 = m0 + (lane >> 4) * 8;
    const int ccol  = n0 + (lane & 15);
    #pragma unroll
    for (int r = 0; r < 8; ++r)
        C[(size_t)(crow0 + r) * DIM + ccol] = acc[r];
}

// ---------------------------------------------------------------- edge scatter
// 64 threads per edge, float4 gather per thread + 4 f32 atomic adds.
__global__ void k_edge_msgs(const float* __restrict__ xw,
                            const long long* __restrict__ ei,
                            const float* __restrict__ ew,
                            const float* __restrict__ dinv,
                            float* __restrict__ agg) {
    int t = blockIdx.x * blockDim.x + threadIdx.x;
    int e = t >> 6;
    if (e >= N_EDGES) return;
    int f  = (t & 63) << 2;
    int s  = (int)ei[e];
    int d  = (int)ei[N_EDGES + e];
    float w = dinv[s] * ew[e] * dinv[d];
    const float4 v = *(const float4*)(xw + (size_t)s * DIM + f);
    float* o = agg + (size_t)d * DIM + f;
    atomicAdd(o + 0, v.x * w);
    atomicAdd(o + 1, v.y * w);
    atomicAdd(o + 2, v.z * w);
    atomicAdd(o + 3, v.w * w);
}

// agg = relu(agg + xw * dinv[i]^2 + b[d])   (in place over agg)
__global__ void k_combine_relu(float* __restrict__ agg,
                               const float* __restrict__ xw,
                               const float* __restrict__ dinv,
                               const float* __restrict__ b) {
    int idx = blockIdx.x * blockDim.x + threadIdx.x;
    if (idx >= N_NODES * DIM) return;
    int i = idx >> 8, d = idx & 255;
    float di = dinv[i];
    float v = agg[idx] + xw[idx] * (di * di) + b[d];
    agg[idx] = v > 0.f ? v : 0.f;
}

// ---------------------------------------------------------------- pooling
__global__ void k_count(const long long* __restrict__ batch, float* __restrict__ cnt) {
    int i = blockIdx.x * blockDim.x + threadIdx.x;
    if (i < N_NODES) atomicAdd(&cnt[(int)batch[i]], 1.0f);
}

// 64 threads per node: float4 chunk -> mean-sum atomicAdd + bitwise atomicMax
__global__ void k_pool(const float* __restrict__ x,
                       const long long* __restrict__ batch,
                       float* __restrict__ msum,
                       float* __restrict__ mmax) {
    int t = blockIdx.x * blockDim.x + threadIdx.x;
    int i = t >> 6;
    if (i >= N_NODES) return;
    int f = (t & 63) << 2;
    int g = (int)batch[i];
    const float4 v = *(const float4*)(x + (size_t)i * DIM + f);
    float* ms = msum + (size_t)g * DIM + f;
    unsigned int* mx = (unsigned int*)(mmax + (size_t)g * DIM + f);
    atomicAdd(ms + 0, v.x); atomicAdd(ms + 1, v.y);
    atomicAdd(ms + 2, v.z); atomicAdd(ms + 3, v.w);
    // post-ReLU values are >= 0, so uint compare == float compare; 0-init
    // matches the reference's (empty/−inf -> 0) semantics.
    atomicMax(mx + 0, __float_as_uint(v.x));
    atomicMax(mx + 1, __float_as_uint(v.y));
    atomicMax(mx + 2, __float_as_uint(v.z));
    atomicMax(mx + 3, __float_as_uint(v.w));
}

// h[g][0:256] = msum/max(cnt,1);  h[g][256:512] = mmax
__global__ void k_pool_final(const float* __restrict__ msum,
                             const float* __restrict__ mmax,
                             const float* __restrict__ cnt,
                             float* __restrict__ h) {
    int idx = blockIdx.x * blockDim.x + threadIdx.x;
    if (idx >= N_GRAPHS * 2 * DIM) return;
    int g = idx / (2 * DIM), d = idx % (2 * DIM);
    float v;
    if (d < DIM) {
        float c = cnt[g];
        v = msum[(size_t)g * DIM + d] / (c > 1.0f ? c : 1.0f);
    } else {
        v = mmax[(size_t)g * DIM + (d - DIM)];
    }
    h[idx] = v;
}

// ---------------------------------------------------------------- tiny FCs
__global__ void k_fc1_relu(const float* __restrict__ h,
                           const float* __restrict__ w,   // [512,256]
                           const float* __restrict__ b,
                           float* __restrict__ out) {     // [128,256]
    int idx = blockIdx.x * blockDim.x + threadIdx.x;
    if (idx >= N_GRAPHS * DIM) return;
    int g = idx >> 8, n = idx & 255;
    float acc = b[n];
    const float* hr = h + (size_t)g * (2 * DIM);
    for (int k = 0; k < 2 * DIM; ++k)
        acc = fmaf(hr[k], w[(size_t)k * DIM + n], acc);
    out[idx] = acc > 0.f ? acc : 0.f;
}

__global__ void k_fc2(const float* __restrict__ h1,
                      const float* __restrict__ w,        // [256,2]
                      const float* __restrict__ b,
                      float* __restrict__ out) {          // [128,2]
    int idx = blockIdx.x * blockDim.x + threadIdx.x;
    if (idx >= N_GRAPHS * N_CLASSES) return;
    int g = idx >> 1, c = idx & 1;
    float acc = b[c];
    const float* hr = h1 + (size_t)g * DIM;
    for (int k = 0; k < DIM; ++k)
        acc = fmaf(hr[k], w[(size_t)k * N_CLASSES + c], acc);
    out[idx] = acc;
}

// ---------------------------------------------------------------- launch
extern "C" void kernel_launch(void* const* d_in, const int* in_sizes, int n_in,
                              void* d_out, int out_size, void* d_ws, size_t ws_size,
                              hipStream_t stream) {
    const float*      emb_x = (const float*)d_in[0];
    const long long*  eidx  = (const long long*)d_in[1];
    const float*      ew    = (const float*)d_in[2];
    const long long*  batch = (const long long*)d_in[3];
    const float* W[3] = { (const float*)d_in[4], (const float*)d_in[6], (const float*)d_in[8] };
    const float* Bv[3] = { (const float*)d_in[5], (const float*)d_in[7], (const float*)d_in[9] };
    const float* fc1_w = (const float*)d_in[10];
    const float* fc1_b = (const float*)d_in[11];
    const float* fc2_w = (const float*)d_in[12];
    const float* fc2_b = (const float*)d_in[13];
    float* out = (float*)d_out;

    // workspace partition (floats)
    float* ws   = (float*)d_ws;
    float* buf0 = ws;                                  // x / agg   [N_NODES*DIM]
    float* buf1 = buf0 + (size_t)N_NODES * DIM;        // xw        [N_NODES*DIM]
    float* dinv = buf1 + (size_t)N_NODES * DIM;        // deg->dinv [N_NODES]
    float* msum = dinv + N_NODES;                      // [128*256]
    float* mmax = msum + N_GRAPHS * DIM;               // [128*256]
    float* cnts = mmax + N_GRAPHS * DIM;               // [128]
    float* h    = cnts + N_GRAPHS;                     // [128*512]
    float* h1   = h + N_GRAPHS * 2 * DIM;              // [128*256]

    const int ND = N_NODES * DIM;

    // degrees -> dinv (graph is identical for all 3 layers)
    k_fill<<<(N_NODES + 255) / 256, 256, 0, stream>>>(dinv, 1.0f, N_NODES);
    k_edge_deg<<<(N_EDGES + 255) / 256, 256, 0, stream>>>(eidx, ew, dinv);
    k_rsqrt_inplace<<<(N_NODES + 255) / 256, 256, 0, stream>>>(dinv, N_NODES);

    const float* x = emb_x;
    for (int l = 0; l < 3; ++l) {
        k_gemm_wmma_f32<<<N_NODES / 16, 512, 0, stream>>>(x, W[l], buf1);
        k_fill<<<(ND + 255) / 256, 256, 0, stream>>>(buf0, 0.0f, ND);
        k_edge_msgs<<<(N_EDGES * 64) / 256, 256, 0, stream>>>(buf1, eidx, ew, dinv, buf0);
        k_combine_relu<<<(ND + 255) / 256, 256, 0, stream>>>(buf0, buf1, dinv, Bv[l]);
        x = buf0;
    }

    // pooling
    k_fill<<<(N_GRAPHS * DIM + 255) / 256, 256, 0, stream>>>(msum, 0.0f, N_GRAPHS * DIM);
    k_fill<<<(N_GRAPHS * DIM + 255) / 256, 256, 0, stream>>>(mmax, 0.0f, N_GRAPHS * DIM);
    k_fill<<<1, 256, 0, stream>>>(cnts, 0.0f, N_GRAPHS);
    k_count<<<(N_NODES + 255) / 256, 256, 0, stream>>>(batch, cnts);
    k_pool<<<(N_NODES * 64) / 256 + 1, 256, 0, stream>>>(buf0, batch, msum, mmax);
    k_pool_final<<<(N_GRAPHS * 2 * DIM + 255) / 256, 256, 0, stream>>>(msum, mmax, cnts, h);

    // heads
    k_fc1_relu<<<(N_GRAPHS * DIM + 255) / 256, 256, 0, stream>>>(h, fc1_w, fc1_b, h1);
    k_fc2<<<1, 256, 0, stream>>>(h1, fc2_w, fc2_b, out);
}